// HCN_58085137711655
// MI455X (gfx1250) — compile-verified
//
#include <hip/hip_runtime.h>
#include <math.h>

typedef float v2f __attribute__((ext_vector_type(2)));
typedef float v8f __attribute__((ext_vector_type(8)));

#define DIM   64
#define NPAD  64   // padded N (relations) stride of score table S

// ---------------------------------------------------------------------------
// Phase 0a: row sums of an [nrows, 64] matrix -> out[nrows]. One wave per row.
// ---------------------------------------------------------------------------
__global__ void rowsum64_kernel(const float* __restrict__ M,
                                float* __restrict__ out, int nrows) {
    int wave = (int)((blockIdx.x * blockDim.x + threadIdx.x) >> 5);
    int lane = threadIdx.x & 31;
    if (wave >= nrows) return;
    const float* row = M + (size_t)wave * DIM;
    float v = row[lane] + row[lane + 32];
    #pragma unroll
    for (int off = 16; off > 0; off >>= 1) v += __shfl_xor(v, off, 32);
    if (lane == 0) out[wave] = v;
}

// ---------------------------------------------------------------------------
// Phase 0b: segment offsets. edge_seg is sorted; row_start[b] = lower_bound(b).
// Launch with B+1 threads so row_start[B] == E.
// ---------------------------------------------------------------------------
__global__ void segstart_kernel(const int* __restrict__ edge_seg, int E,
                                int* __restrict__ row_start, int n) {
    int b = blockIdx.x * blockDim.x + threadIdx.x;
    if (b >= n) return;
    int lo = 0, hi = E;
    while (lo < hi) {
        int mid = (lo + hi) >> 1;
        if (edge_seg[mid] < b) lo = mid + 1; else hi = mid;
    }
    row_start[b] = lo;
}

// ---------------------------------------------------------------------------
// Phase 1: score table S[NHpad, 64] = H(NHx64) * R^T(64xNR) via
// V_WMMA_F32_16X16X4_F32. One wave per 16x16 C tile; blockDim=128 -> 4 waves
// cover the 4 N-tiles (NR=60 padded to 64); gridDim.x = ceil(NH/16).
//
// No bounds predication: out-of-range rows/cols are CLAMPED into bounds, so
// loads are unconditional (v2f = global_load_b64) and can all be issued
// before the WMMA chain. Garbage products land only in padded S rows (>= NH)
// and cols (>= NR), which phase 2 never reads (head < NH, rel < NR).
//
// A layout (16x4 f32): lanes 0-15 hold M=0..15 {v0:K=0, v1:K=1},
//                      lanes 16-31 hold M=0..15 {v0:K=2, v1:K=3}.
// B layout mirrors over N. C/D: VGPR r holds M = r + 8*(lane>=16), N=lane%16.
// ---------------------------------------------------------------------------
__global__ void score_gemm_kernel(const float* __restrict__ H,   // [NH,64]
                                  const float* __restrict__ R,   // [NR,64]
                                  float* __restrict__ S,         // [NHpad,64]
                                  int NH, int NR) {
    const int lane  = threadIdx.x & 31;
    const int nTile = threadIdx.x >> 5;        // 0..3
    const int mTile = blockIdx.x;
    const int half  = lane >> 4;
    const int l16   = lane & 15;

    const int mRow = min(mTile * 16 + l16, NH - 1);  // clamped A row
    const int nCol = min(nTile * 16 + l16, NR - 1);  // clamped B column

    const float* aPtr = H + (size_t)mRow * DIM + 2 * half;
    const float* bPtr = R + (size_t)nCol * DIM + 2 * half;

    // Preload all K-slices (8-byte aligned: 16*kk + 8*half byte offsets).
    v2f a[DIM / 4], b[DIM / 4];
    #pragma unroll
    for (int kk = 0; kk < DIM / 4; ++kk) {
        a[kk] = *reinterpret_cast<const v2f*>(aPtr + kk * 4);
        b[kk] = *reinterpret_cast<const v2f*>(bPtr + kk * 4);
    }

    v8f c = {};
    #pragma unroll
    for (int kk = 0; kk < DIM / 4; ++kk) {
        // (neg_a, A, neg_b, B, c_mod, C, reuse_a, reuse_b)
        c = __builtin_amdgcn_wmma_f32_16x16x4_f32(false, a[kk], false, b[kk],
                                                  (short)0, c, false, false);
    }

    #pragma unroll
    for (int r = 0; r < 8; ++r) {
        const int row = mTile * 16 + r + 8 * half;
        const int col = nTile * 16 + l16;      // always < NPAD
        S[(size_t)row * NPAD + col] = c[r];
    }
}

// ---------------------------------------------------------------------------
// Phase 2: one wave32 per batch row. Online segment softmax over the row's
// contiguous edge range, combining (m, sum_e, sum_e*val) across lanes, then
// broadcast the scalar into the 64-wide output row (float2 per lane).
// ---------------------------------------------------------------------------
__global__ void edge_softmax_kernel(const int*   __restrict__ h,
                                    const int*   __restrict__ edge_rel,
                                    const int*   __restrict__ edge_tail,
                                    const float* __restrict__ S,
                                    const float* __restrict__ tsum,
                                    const float* __restrict__ rsum,
                                    const int*   __restrict__ row_start,
                                    float*       __restrict__ out, int B) {
    int b    = (int)((blockIdx.x * blockDim.x + threadIdx.x) >> 5);
    int lane = threadIdx.x & 31;
    if (b >= B) return;

    const float* sRow = S + (size_t)h[b] * NPAD;
    const int start = row_start[b];
    const int end   = row_start[b + 1];

    float m = -INFINITY, s = 0.f, c = 0.f;
    for (int e = start + lane; e < end; e += 32) {
        const int   rel = edge_rel[e];
        const float sc  = sRow[rel];
        const float val = tsum[edge_tail[e]] - rsum[rel];
        const float nm    = fmaxf(m, sc);
        const float t     = __expf(sc - nm);
        const float scale = (s > 0.f) ? __expf(m - nm) : 0.f;
        s = s * scale + t;
        c = c * scale + t * val;
        m = nm;
    }

    #pragma unroll
    for (int off = 16; off > 0; off >>= 1) {
        const float m2 = __shfl_xor(m, off, 32);
        const float s2 = __shfl_xor(s, off, 32);
        const float c2 = __shfl_xor(c, off, 32);
        const float nm = fmaxf(m, m2);
        const float a1 = (s  > 0.f) ? __expf(m  - nm) : 0.f;
        const float a2 = (s2 > 0.f) ? __expf(m2 - nm) : 0.f;
        s = s * a1 + s2 * a2;
        c = c * a1 + c2 * a2;
        m = nm;
    }

    const float vc = c / ((s > 0.f) ? s : 1.f);
    float2 o; o.x = vc; o.y = vc;
    *reinterpret_cast<float2*>(out + (size_t)b * DIM + lane * 2) = o;
}

// ---------------------------------------------------------------------------
extern "C" void kernel_launch(void* const* d_in, const int* in_sizes, int n_in,
                              void* d_out, int out_size, void* d_ws, size_t ws_size,
                              hipStream_t stream) {
    const int*   h         = (const int*)  d_in[0];
    const int*   edge_seg  = (const int*)  d_in[1];
    const int*   edge_rel  = (const int*)  d_in[2];
    const int*   edge_tail = (const int*)  d_in[3];
    const float* H_emb     = (const float*)d_in[4];
    const float* R_emb     = (const float*)d_in[5];
    const float* T_emb     = (const float*)d_in[6];

    const int B  = in_sizes[0];
    const int E  = in_sizes[1];
    const int NH = in_sizes[4] / DIM;   // 3846
    const int NR = in_sizes[5] / DIM;   // 60
    const int NT = in_sizes[6] / DIM;   // 9366

    const int mTiles = (NH + 15) / 16;
    const int NHpad  = mTiles * 16;

    // Workspace layout (all f32 / i32, ~1.2 MB total)
    float* S         = (float*)d_ws;                 // [NHpad, 64]
    float* tsum      = S + (size_t)NHpad * NPAD;     // [NT]
    float* rsum      = tsum + NT;                    // [NR]
    int*   row_start = (int*)(rsum + NR);            // [B+1]

    // Phase 0: row sums + segment offsets
    {
        int thr = NT * 32;
        rowsum64_kernel<<<(thr + 255) / 256, 256, 0, stream>>>(T_emb, tsum, NT);
        thr = NR * 32;
        rowsum64_kernel<<<(thr + 255) / 256, 256, 0, stream>>>(R_emb, rsum, NR);
        segstart_kernel<<<(B + 1 + 255) / 256, 256, 0, stream>>>(edge_seg, E,
                                                                 row_start, B + 1);
    }

    // Phase 1: WMMA score table  S = H x R^T   (N padded to 64 -> 4 waves/block)
    score_gemm_kernel<<<mTiles, 128, 0, stream>>>(H_emb, R_emb, S, NH, NR);

    // Phase 2: segment softmax + scalar broadcast, one wave per batch row
    {
        long long thr = (long long)B * 32;
        edge_softmax_kernel<<<(int)((thr + 255) / 256), 256, 0, stream>>>(
            h, edge_rel, edge_tail, S, tsum, rsum, row_start,
            (float*)d_out, B);
    }
}